// DVGGA_67551245631646
// MI455X (gfx1250) — compile-verified
//
#include <hip/hip_runtime.h>
#include <hip/hip_bf16.h>

typedef __attribute__((ext_vector_type(2))) float v2f;
typedef __attribute__((ext_vector_type(8))) float v8f;

// Problem constants
constexpr int GG  = 2048;   // graphs
constexpr int NN  = 128;    // nodes per graph
constexpr int EGN = 1024;   // edges per graph
constexpr int FF  = 64;     // input feature dim
constexpr int FF1 = 256;    // gcn1 / fc1 dim
constexpr int KC  = 16;     // pool clusters
constexpr int EE  = 32768;  // pos/neg edges

// One 16x16x4 fp32 WMMA step.  A: LDS/global row-major (lda), B: global row-major (ldb).
__device__ inline v8f wmma_step(const float* __restrict__ A, int lda,
                                const float* __restrict__ B, int ldb,
                                int m0, int n0, int k0, int lm, int lh, v8f acc) {
    const float* ap = A + (size_t)(m0 + lm) * lda + k0 + 2 * lh;
    v2f a = { ap[0], ap[1] };
    const float* bp = B + (size_t)(k0 + 2 * lh) * ldb + n0 + lm;
    v2f b = { bp[0], bp[ldb] };
    return __builtin_amdgcn_wmma_f32_16x16x4_f32(false, a, false, b, (short)0, acc, false, false);
}

// CDNA5 async memory->LDS copy (16B per lane), tracked by ASYNCcnt.
__device__ inline void async_b128(void* lds_ptr, const void* gptr) {
    unsigned lds_off = (unsigned)(size_t)lds_ptr;   // LDS address is the low 32 bits of the pointer
    asm volatile("global_load_async_to_lds_b128 %0, %1, off"
                 :: "v"(lds_off), "v"(gptr) : "memory");
}
__device__ inline void async_wait0() {
    asm volatile("s_wait_asynccnt 0x0" ::: "memory");
}

// ---------------------------------------------------------------------------
// Per-graph kernel: one workgroup (8 waves) per graph.
// ---------------------------------------------------------------------------
__global__ __launch_bounds__(256) void k_graph(
    const float* __restrict__ feats,   // G x 128 x 64
    const int*   __restrict__ edges,   // G x 2 x 1024
    const float* __restrict__ w1,  const float* __restrict__ b1,   // 64x256, 256
    const float* __restrict__ wf1, const float* __restrict__ bf1,  // 256x256, 256
    const float* __restrict__ wf2, const float* __restrict__ bf2,  // 256x16, 16
    float* __restrict__ ge,            // G x 256
    float* __restrict__ accPen)
{
    __shared__ float sFeat[NN * FF];       // 32KB: feat, later a1 scratch (128x64)
    __shared__ float sX1[NN * 128];        // 64KB: x1 half; later S (128x16) + T (128x16)
    __shared__ float sX2[NN * FF1];        // 128KB: propagated features
    __shared__ float sDvD[NN], sDvS[NN];
    __shared__ int   sSrc[EGN], sDst[EGN];
    __shared__ float sNA[KC * KC];
    __shared__ float sDg[KC];

    const int g = blockIdx.x;
    const int tid = threadIdx.x;
    const int w = tid >> 5, lane = tid & 31;
    const int lm = lane & 15, lh = lane >> 4;

    const float* feat = feats + (size_t)g * NN * FF;
    const int* esrc = edges + (size_t)g * 2 * EGN;
    const int* edst = esrc + EGN;

    // Warm L2 for the big reused weight (256KB, shared by all 2048 blocks).
    __builtin_prefetch(wf1 + tid * 256, 0, 3);

    // Async-stage feat into LDS (memory -> LDS, no VGPR round trip).
    for (int i = tid; i < NN * FF / 4; i += 256)
        async_b128(sFeat + 4 * i, feat + 4 * i);
    for (int e = tid; e < EGN; e += 256) { sSrc[e] = esrc[e]; sDst[e] = edst[e]; }
    if (tid < NN) { sDvD[tid] = 1.0f; sDvS[tid] = 0.0f; }
    async_wait0();
    __syncthreads();
    for (int e = tid; e < EGN; e += 256) {
        atomicAdd(&sDvD[sDst[e]], 1.0f);
        atomicAdd(&sDvS[sSrc[e]], 1.0f);
    }
    __syncthreads();
    if (tid < NN) {
        sDvD[tid] = rsqrtf(sDvD[tid]);                       // always >= 1
        float d = sDvS[tid];
        sDvS[tid] = d > 0.f ? rsqrtf(d) : 0.f;
    }
    __syncthreads();

    // x1 = feat @ w1, propagated into x2, in two 128-column halves.
    for (int half = 0; half < 2; ++half) {
        // GEMM: 8 Mtiles x 8 Ntiles, K=64
        for (int t = w; t < 64; t += 8) {
            int mi = t >> 3, ni = t & 7;
            v8f acc = {};
            for (int k0 = 0; k0 < FF; k0 += 4)
                acc = wmma_step(sFeat, FF, w1, FF1, mi * 16, half * 128 + ni * 16, k0, lm, lh, acc);
            #pragma unroll
            for (int v = 0; v < 8; ++v)
                sX1[(mi * 16 + 8 * lh + v) * 128 + ni * 16 + lm] = acc[v];
        }
        __syncthreads();
        // self loops: x2[n] = dinv[n]^2 * x1[n]
        for (int i = tid; i < NN * 128; i += 256) {
            int n = i >> 7, f = i & 127;
            float dv = sDvD[n];
            sX2[n * FF1 + half * 128 + f] = dv * dv * sX1[i];
        }
        __syncthreads();
        // edge scatter: x2[dst] += dinv[src]*dinv[dst] * x1[src]
        for (int e = tid; e < EGN; e += 256) {
            int s_ = sSrc[e], d_ = sDst[e];
            float c = sDvD[s_] * sDvD[d_];
            const float* xs = sX1 + s_ * 128;
            float* xd = sX2 + d_ * FF1 + half * 128;
            for (int f = 0; f < 128; ++f)
                atomicAdd(&xd[f], c * xs[f]);
        }
        __syncthreads();
    }
    for (int i = tid; i < NN * FF1; i += 256)   // + gcn1_b
        sX2[i] += b1[i & 255];
    __syncthreads();

    // a1 = tanh(x2 @ wf1 + bf1) tiled by 64 cols; logits += a1_tile @ wf2_tile.
    v8f accL = {};
    for (int ng = 0; ng < 4; ++ng) {
        for (int t = w; t < 32; t += 8) {
            int mi = t >> 2, nj = t & 3;
            int n0 = ng * 64 + nj * 16;
            v8f acc = {};
            for (int k0 = 0; k0 < FF1; k0 += 4)
                acc = wmma_step(sX2, FF1, wf1, FF1, mi * 16, n0, k0, lm, lh, acc);
            float bias = bf1[n0 + lm];
            #pragma unroll
            for (int v = 0; v < 8; ++v)
                sFeat[(mi * 16 + 8 * lh + v) * 64 + nj * 16 + lm] = tanhf(acc[v] + bias);
        }
        __syncthreads();
        // logits partial: wave w owns M-tile w, K = 64 cols of a1 scratch
        for (int k0 = 0; k0 < 64; k0 += 4) {
            const float* ap = sFeat + (w * 16 + lm) * 64 + k0 + 2 * lh;
            v2f a = { ap[0], ap[1] };
            const float* bp = wf2 + (ng * 64 + k0 + 2 * lh) * KC + lm;
            v2f b = { bp[0], bp[KC] };
            accL = __builtin_amdgcn_wmma_f32_16x16x4_f32(false, a, false, b, (short)0, accL, false, false);
        }
        __syncthreads();
    }
    float* sS = sX1;                 // 128x16
    float* sT = sX1 + NN * KC;       // 128x16
    {
        float bias2 = bf2[lm];
        #pragma unroll
        for (int v = 0; v < 8; ++v)
            sS[(w * 16 + 8 * lh + v) * KC + lm] = accL[v] + bias2;
    }
    __syncthreads();

    // Row softmax over 16 cols; zero T.
    if (tid < NN) {
        float* r = sS + tid * KC;
        float mx = r[0];
        for (int j = 1; j < KC; ++j) mx = fmaxf(mx, r[j]);
        float s = 0.f;
        for (int j = 0; j < KC; ++j) { float e0 = __expf(r[j] - mx); r[j] = e0; s += e0; }
        float inv = 1.f / s;
        for (int j = 0; j < KC; ++j) r[j] *= inv;
    }
    for (int i = tid; i < NN * KC; i += 256) sT[i] = 0.f;
    __syncthreads();
    // T[src] += dinvS[src]*dinvS[dst] * S[dst]
    for (int e = tid; e < EGN; e += 256) {
        int s_ = sSrc[e], d_ = sDst[e];
        float c = sDvS[s_] * sDvS[d_];
        const float* sd = sS + d_ * KC;
        float* ts = sT + s_ * KC;
        for (int f = 0; f < KC; ++f)
            atomicAdd(&ts[f], c * sd[f]);
    }
    __syncthreads();
    for (int i = tid; i < NN * KC; i += 256) sT[i] = sS[i] - sT[i];   // LS
    __syncthreads();
    // new_adj[i][j] = sum_n S[n][i]*LS[n][j]
    {
        int i = tid >> 4, j = tid & 15;
        float s = 0.f;
        for (int n = 0; n < NN; ++n) s += sS[n * KC + i] * sT[n * KC + j];
        sNA[tid] = s;
    }
    __syncthreads();
    if (tid < KC) {  // diag of row-normalized new_adj
        float s = 0.f;
        for (int j = 0; j < KC; ++j) s += fabsf(sNA[tid * KC + j]);
        sDg[tid] = sNA[tid * KC + tid] / fmaxf(s, 1e-12f);
    }
    // ge[f] = (1/16) * sum_n x2[n][f]  (softmax rows sum to 1)
    {
        float s = 0.f;
        for (int n = 0; n < NN; ++n) s += sX2[n * FF1 + tid];
        ge[(size_t)g * FF1 + tid] = s * (1.f / 16.f);
    }
    __syncthreads();
    if (tid == 0) {
        float s = 0.f;
        for (int i = 0; i < KC; ++i)
            for (int j = 0; j < KC; ++j) {
                float v0 = sDg[j] - (i == j ? 1.f : 0.f);
                s += v0 * v0;
            }
        atomicAdd(accPen, s * (1.f / (KC * KC)));
    }
}

// ---------------------------------------------------------------------------
// Generic WMMA GEMM: C(MxN) = A(MxK) @ B(KxN), all row-major fp32.
// 8 tiles per block (one 16x16 tile per wave).
// ---------------------------------------------------------------------------
__global__ __launch_bounds__(256) void k_gemm(
    const float* __restrict__ A, int lda,
    const float* __restrict__ B, int ldb,
    float* __restrict__ C, int ldc,
    int Nt, int nTiles, int K)
{
    int tid = threadIdx.x, w = tid >> 5, lane = tid & 31;
    int lm = lane & 15, lh = lane >> 4;
    __builtin_prefetch(B + tid * 256, 0, 3);
    int tile = blockIdx.x * 8 + w;
    if (tile >= nTiles) return;          // wave-uniform
    int mi = tile / Nt, ni = tile % Nt;
    v8f acc = {};
    for (int k0 = 0; k0 < K; k0 += 4)
        acc = wmma_step(A, lda, B, ldb, mi * 16, ni * 16, k0, lm, lh, acc);
    #pragma unroll
    for (int v = 0; v < 8; ++v)
        C[(size_t)(mi * 16 + 8 * lh + v) * ldc + ni * 16 + lm] = acc[v];
}

// ---------------------------------------------------------------------------
// Graph-level GCN prop helpers (n = 2048, edges = pos_edges).
// ---------------------------------------------------------------------------
__global__ void k_init(float* deg, float* accums) {
    int i = blockIdx.x * blockDim.x + threadIdx.x;
    if (i < GG) deg[i] = 1.0f;           // self-loop contribution to degree
    if (i < 8)  accums[i] = 0.f;
}
__global__ void k_deg(const int* __restrict__ pe, float* deg) {
    int e = blockIdx.x * blockDim.x + threadIdx.x;
    if (e < EE) atomicAdd(&deg[pe[2 * e + 1]], 1.0f);
}
__global__ void k_dinv(float* deg) {
    int i = blockIdx.x * blockDim.x + threadIdx.x;
    if (i < GG) deg[i] = rsqrtf(deg[i]);
}
__global__ void k_self(const float* __restrict__ in, const float* __restrict__ dinv,
                       float* __restrict__ out, int F) {
    int i = blockIdx.x * blockDim.x + threadIdx.x;
    int n = i / F;
    float d = dinv[n];
    out[i] = d * d * in[i];
}
__global__ void k_scatter(const float* __restrict__ in, const int* __restrict__ pe,
                          const float* __restrict__ dinv, float* __restrict__ out, int F) {
    int e = blockIdx.x;
    int f = threadIdx.x;                 // blockDim == F
    int s = pe[2 * e], d = pe[2 * e + 1];
    float c = dinv[s] * dinv[d];
    atomicAdd(&out[(size_t)d * F + f], c * in[(size_t)s * F + f]);
}
__global__ void k_relu_bias(float* x, const float* __restrict__ b, int F) {
    int i = blockIdx.x * blockDim.x + threadIdx.x;
    x[i] = fmaxf(x[i] + b[i & (F - 1)], 0.f);
}

// Reparameterize + build z = [emb | mu + eps*exp(logstd)] and accumulate KL sum.
__global__ __launch_bounds__(256) void k_z(
    const float* __restrict__ muA, const float* __restrict__ lsA,
    const float* __restrict__ mu_b, const float* __restrict__ ls_b,
    const float* __restrict__ eps, const float* __restrict__ emb,
    float* __restrict__ z, float* __restrict__ accums)
{
    __shared__ float red[256];
    int i = blockIdx.x * 256 + threadIdx.x;      // over 2048*128
    int n = i >> 7, f = i & 127;
    float m = muA[i] + mu_b[f];
    float l = fminf(lsA[i] + ls_b[f], 10.f);
    z[(size_t)n * 256 + 128 + f] = m + eps[i] * __expf(l);
    z[(size_t)n * 256 + f] = emb[i];
    red[threadIdx.x] = 1.f + 2.f * l - m * m - __expf(2.f * l);
    __syncthreads();
    for (int s = 128; s > 0; s >>= 1) {
        if (threadIdx.x < s) red[threadIdx.x] += red[threadIdx.x + s];
        __syncthreads();
    }
    if (threadIdx.x == 0) atomicAdd(&accums[1], red[0]);
}

// ---------------------------------------------------------------------------
// Edge scoring: 64 edges per block; fx/fy tiles via WMMA, dots accumulated.
// ---------------------------------------------------------------------------
__global__ __launch_bounds__(256) void k_score(
    const float* __restrict__ z, const int* __restrict__ ei,
    const float* __restrict__ w1, const float* __restrict__ b1v,
    const float* __restrict__ w2, const float* __restrict__ b2v,
    float* __restrict__ pred, float* __restrict__ accLog, int isPos)
{
    __shared__ float sZx[64 * 256];
    __shared__ float sZy[64 * 256];
    __shared__ float sDot[64];
    __shared__ float sRed[256];
    int tid = threadIdx.x, w = tid >> 5, lane = tid & 31;
    int lm = lane & 15, lh = lane >> 4;
    int e0 = blockIdx.x * 64;

    __builtin_prefetch(w1 + tid * 256, 0, 3);
    __builtin_prefetch(w2 + tid * 256, 0, 3);

    // Async gather of z rows directly into LDS.
    for (int i = tid; i < 64 * 64; i += 256) {
        int r = i >> 6, c4 = i & 63;
        int ex = ei[2 * (e0 + r)];
        int ey = ei[2 * (e0 + r) + 1];
        async_b128(sZx + r * 256 + 4 * c4, z + (size_t)ex * 256 + 4 * c4);
        async_b128(sZy + r * 256 + 4 * c4, z + (size_t)ey * 256 + 4 * c4);
    }
    if (tid < 64) sDot[tid] = 0.f;
    async_wait0();
    __syncthreads();

    // 4 Mtiles x 16 Ntiles of (fx, fy) pairs; K = 256
    for (int t = w; t < 64; t += 8) {
        int mi = t >> 4, ni = t & 15;
        v8f ax = {}, ay = {};
        for (int k0 = 0; k0 < 256; k0 += 4) {
            ax = wmma_step(sZx, 256, w1, 256, mi * 16, ni * 16, k0, lm, lh, ax);
            ay = wmma_step(sZy, 256, w2, 256, mi * 16, ni * 16, k0, lm, lh, ay);
        }
        float bx = b1v[ni * 16 + lm], by = b2v[ni * 16 + lm];
        #pragma unroll
        for (int v = 0; v < 8; ++v)
            atomicAdd(&sDot[mi * 16 + 8 * lh + v], (ax[v] + bx) * (ay[v] + by));
    }
    __syncthreads();

    float lg = 0.f;
    if (tid < 64) {
        float d = sDot[tid];
        float pr = 1.f / (1.f + __expf(d));      // sigmoid(-dot)
        pred[e0 + tid] = pr;
        lg = isPos ? __logf(pr + 1e-15f) : __logf(1.f - pr + 1e-15f);
    }
    sRed[tid] = lg;
    __syncthreads();
    for (int s = 128; s > 0; s >>= 1) {
        if (tid < s) sRed[tid] += sRed[tid + s];
        __syncthreads();
    }
    if (tid == 0) atomicAdd(accLog, sRed[0]);
}

__global__ void k_final(const float* __restrict__ accums, float* __restrict__ out) {
    if (threadIdx.x == 0) {
        float rec = -(accums[2] / (float)EE) - (accums[3] / (float)EE);
        float kl  = -0.5f * accums[1] / ((float)GG * (float)GG);
        out[0] = rec + kl;
        out[1] = accums[0] / (float)GG;
    }
}

// ---------------------------------------------------------------------------
extern "C" void kernel_launch(void* const* d_in, const int* in_sizes, int n_in,
                              void* d_out, int out_size, void* d_ws, size_t ws_size,
                              hipStream_t stream) {
    const float* features = (const float*)d_in[0];
    const float* eps      = (const float*)d_in[1];
    const float* gcn1_w   = (const float*)d_in[2];
    const float* gcn1_b   = (const float*)d_in[3];
    const float* fc1_w    = (const float*)d_in[4];
    const float* fc1_b    = (const float*)d_in[5];
    const float* fc2_w    = (const float*)d_in[6];
    const float* fc2_b    = (const float*)d_in[7];
    const float* conv1_w  = (const float*)d_in[8];
    const float* conv1_b  = (const float*)d_in[9];
    const float* mu_w     = (const float*)d_in[10];
    const float* mu_b     = (const float*)d_in[11];
    const float* ls_w     = (const float*)d_in[12];
    const float* ls_b     = (const float*)d_in[13];
    const float* emb      = (const float*)d_in[14];
    const float* lin1_w   = (const float*)d_in[15];
    const float* lin1_b   = (const float*)d_in[16];
    const float* lin2_w   = (const float*)d_in[17];
    const float* lin2_b   = (const float*)d_in[18];
    const int*   edges    = (const int*)d_in[19];
    const int*   pos_e    = (const int*)d_in[20];
    const int*   neg_e    = (const int*)d_in[21];
    float* out = (float*)d_out;

    float* ws    = (float*)d_ws;
    float* ge    = ws;                 // 2048*256
    float* hacc  = ws + 524288;        // 2048*256 (h in place)
    float* hpre  = ws + 1048576;       // 2048*256
    float* mupre = ws + 1572864;       // 2048*128
    float* lspre = ws + 1835008;       // 2048*128
    float* muacc = ws + 2097152;       // 2048*128
    float* lsacc = ws + 2359296;       // 2048*128
    float* zbuf  = ws + 2621440;       // 2048*256
    float* deg   = ws + 3145728;       // 2048 (becomes dinv)
    float* acc   = ws + 3147776;       // 8 accumulators: pen, kl, poslog, neglog

    k_init<<<8, 256, 0, stream>>>(deg, acc);

    // Per-graph pipeline (pooling + embedding + penalty)
    k_graph<<<GG, 256, 0, stream>>>(features, edges, gcn1_w, gcn1_b,
                                    fc1_w, fc1_b, fc2_w, fc2_b, ge, &acc[0]);

    // Graph-level degree / dinv over pos_edges (+ self loops)
    k_deg<<<EE / 256, 256, 0, stream>>>(pos_e, deg);
    k_dinv<<<8, 256, 0, stream>>>(deg);

    // h = relu(gcn_prop(ge @ conv1_w) + conv1_b)
    k_gemm<<<256, 256, 0, stream>>>(ge, 256, conv1_w, 256, hpre, 256, 16, 2048, 256);
    k_self<<<GG * 256 / 256, 256, 0, stream>>>(hpre, deg, hacc, 256);
    k_scatter<<<EE, 256, 0, stream>>>(hpre, pos_e, deg, hacc, 256);
    k_relu_bias<<<GG * 256 / 256, 256, 0, stream>>>(hacc, conv1_b, 256);

    // mu / logstd projections + props
    k_gemm<<<128, 256, 0, stream>>>(hacc, 256, mu_w, 128, mupre, 128, 8, 1024, 256);
    k_gemm<<<128, 256, 0, stream>>>(hacc, 256, ls_w, 128, lspre, 128, 8, 1024, 256);
    k_self<<<GG * 128 / 256, 256, 0, stream>>>(mupre, deg, muacc, 128);
    k_scatter<<<EE, 128, 0, stream>>>(mupre, pos_e, deg, muacc, 128);
    k_self<<<GG * 128 / 256, 256, 0, stream>>>(lspre, deg, lsacc, 128);
    k_scatter<<<EE, 128, 0, stream>>>(lspre, pos_e, deg, lsacc, 128);

    // z = [emb | mu + eps*exp(logstd)] and KL accumulation
    k_z<<<GG * 128 / 256, 256, 0, stream>>>(muacc, lsacc, mu_b, ls_b, eps, emb, zbuf, acc);

    // Edge scoring (predictions + log sums)
    k_score<<<EE / 64, 256, 0, stream>>>(zbuf, pos_e, lin1_w, lin1_b, lin2_w, lin2_b,
                                         out + 2, &acc[2], 1);
    k_score<<<EE / 64, 256, 0, stream>>>(zbuf, neg_e, lin1_w, lin1_b, lin2_w, lin2_b,
                                         out + 2 + EE, &acc[3], 0);

    k_final<<<1, 64, 0, stream>>>(acc, out);
}